// NM_block_7095285973250
// MI455X (gfx1250) — compile-verified
//
#include <hip/hip_runtime.h>
#include <hip/hip_bf16.h>

typedef float v2f __attribute__((ext_vector_type(2)));
typedef float v8f __attribute__((ext_vector_type(8)));

__device__ __forceinline__ v8f wmma4(v2f a, v2f b, v8f c) {
  // D = A(16x4,f32) x B(4x16,f32) + C(16x16,f32)  -> v_wmma_f32_16x16x4_f32
  return __builtin_amdgcn_wmma_f32_16x16x4_f32(false, a, false, b, (short)0, c, false, false);
}

#define BB 8
#define CC 128
#define NN 2048
#define CP 132           // channels padded (129 -> 132) for K%4==0
#define KNB 20           // neighbors
#define CNT0 327680.0f   // B*N*K for bn0 stats

// ---------------- phase 1: norms, column sums, feature build -----------------

__global__ void k_norms(const float* __restrict__ data, float* __restrict__ nrm) {
  int p = blockIdx.x * 256 + threadIdx.x;          // 0..16383
  int b = p >> 11, n = p & 2047;
  const float* dp = data + (size_t)b * CC * NN + n;
  float s = 0.f;
  for (int c = 0; c < CC; ++c) { float v = dp[(size_t)c * NN]; s += v * v; }
  nrm[p] = sqrtf(s) + 1e-5f;
}

__global__ void k_colsum(const float* __restrict__ data, const float* __restrict__ nrm,
                         float* __restrict__ s) {
  int b = blockIdx.x >> 7, c = blockIdx.x & 127;
  const float* dp = data + (size_t)(b * CC + c) * NN;
  const float* np = nrm + b * NN;
  __shared__ float red[256];
  float acc = 0.f;
  for (int n = threadIdx.x; n < NN; n += 256) acc += dp[n] / np[n];
  red[threadIdx.x] = acc; __syncthreads();
  for (int st = 128; st > 0; st >>= 1) {
    if (threadIdx.x < st) red[threadIdx.x] += red[threadIdx.x + st];
    __syncthreads();
  }
  if (threadIdx.x == 0) s[blockIdx.x] = red[0];
}

__global__ void k_feat(const float* __restrict__ data, const float* __restrict__ s,
                       float* __restrict__ featP, float* __restrict__ q) {
  int p = blockIdx.x * 256 + threadIdx.x;
  int b = p >> 11, n = p & 2047;
  const float* dp = data + (size_t)b * CC * NN + n;
  const float* sp = s + b * CC;
  float* fp = featP + (size_t)b * CP * NN + n;
  float sd2 = 0.f, sds = 0.f;
  for (int c = 0; c < CC; ++c) {
    float v = dp[(size_t)c * NN];
    sd2 += v * v; sds += v * sp[c];
    fp[(size_t)c * NN] = v;
  }
  float dot = sds / (sqrtf(sd2) + 1e-5f);
  float D = (dot - 1.f) / (float)NN;
  float Dp = fmaxf(tanhf(D), 0.f);
  fp[(size_t)128 * NN] = Dp;
  fp[(size_t)129 * NN] = 0.f; fp[(size_t)130 * NN] = 0.f; fp[(size_t)131 * NN] = 0.f;
  q[p] = sd2 + Dp * Dp;
}

// ---------------- phase 2: kNN via f32 WMMA Gram + chunked top-20 -----------

__global__ __launch_bounds__(128) void k_knn(const float* __restrict__ featP,
                                             const float* __restrict__ q,
                                             int* __restrict__ idxb) {
  __shared__ float dist[16 * 512];
  __shared__ float rv[128]; __shared__ int ri[128];
  __shared__ float candv[16 * 20]; __shared__ int candi[16 * 20];
  __shared__ float runv[16 * 20];  __shared__ int runi[16 * 20];
  __shared__ float qrow[16];
  int b = blockIdx.y, rowBase = blockIdx.x * 16;
  int tid = threadIdx.x, lane = tid & 31, wave = tid >> 5;
  int l15 = lane & 15, khalf = (lane >> 4) * 2, rofs = (lane < 16) ? 0 : 8;
  const float* fb = featP + (size_t)b * CP * NN;
  const float* qb = q + b * NN;
  if (tid < 16) qrow[tid] = qb[rowBase + tid];
  __syncthreads();
  int rowI = rowBase + l15;
  for (int chunk = 0; chunk < 4; ++chunk) {
    int cb0 = chunk * 512;
    for (int ct = wave; ct < 32; ct += 4) {
      int colI = cb0 + ct * 16 + l15;
      v8f acc = {};
      for (int kc = 0; kc < 33; ++kc) {
        int kp = kc * 4 + khalf;
        v2f a, bm;
        a.x = fb[(size_t)kp * NN + rowI];       a.y = fb[(size_t)(kp + 1) * NN + rowI];
        bm.x = fb[(size_t)kp * NN + colI];      bm.y = fb[(size_t)(kp + 1) * NN + colI];
        acc = wmma4(a, bm, acc);
      }
      float qc = qb[colI];
      for (int r = 0; r < 8; ++r) {
        int row = r + rofs;
        float v = 2.f * acc[r] - qrow[row] - qc;
        if (rowBase + row == colI) v = -3.0e38f;   // exclude self
        dist[row * 512 + (colI - cb0)] = v;
      }
    }
    __syncthreads();
    int rowId = tid >> 3, sub = tid & 7;
    for (int sel = 0; sel < 20; ++sel) {
      float best = -3.39e38f; int bidx = sub * 64;
      for (int i = sub * 64; i < sub * 64 + 64; ++i) {
        float v = dist[rowId * 512 + i];
        if (v > best) { best = v; bidx = i; }
      }
      rv[tid] = best; ri[tid] = bidx;
      __syncthreads();
      if (sub == 0) {
        float bv = rv[rowId * 8]; int bi = ri[rowId * 8];
        for (int j = 1; j < 8; ++j)
          if (rv[rowId * 8 + j] > bv) { bv = rv[rowId * 8 + j]; bi = ri[rowId * 8 + j]; }
        candv[rowId * 20 + sel] = bv; candi[rowId * 20 + sel] = cb0 + bi;
        dist[rowId * 512 + bi] = -3.4e38f;
      }
      __syncthreads();
    }
    if (tid < 16) {
      if (chunk == 0) {
        for (int j = 0; j < 20; ++j) { runv[tid * 20 + j] = candv[tid * 20 + j]; runi[tid * 20 + j] = candi[tid * 20 + j]; }
      } else {  // merge two descending-sorted 20-lists, keep top 20
        float tv[20]; int ti[20]; int i = 0, j = 0;
        for (int s2 = 0; s2 < 20; ++s2) {
          if (runv[tid * 20 + i] >= candv[tid * 20 + j]) { tv[s2] = runv[tid * 20 + i]; ti[s2] = runi[tid * 20 + i]; ++i; }
          else { tv[s2] = candv[tid * 20 + j]; ti[s2] = candi[tid * 20 + j]; ++j; }
        }
        for (int s2 = 0; s2 < 20; ++s2) { runv[tid * 20 + s2] = tv[s2]; runi[tid * 20 + s2] = ti[s2]; }
      }
    }
    __syncthreads();
  }
  if (tid < 16) {
    int* op = idxb + ((size_t)b * NN + rowBase + tid) * KNB;
    for (int j = 0; j < 20; ++j) op[j] = runi[tid * 20 + j];
  }
}

// ---------------- phase 3: conv0 (edge conv) + bn0 stats --------------------

__global__ void k_prepw(const float* __restrict__ w0, float* __restrict__ wsum,
                        float* __restrict__ w2) {
  int i = blockIdx.x * 256 + threadIdx.x;
  if (i >= CC * CP) return;
  int o = i / CP, c = i % CP;
  float a = 0.f, s = 0.f;
  if (c < 129) { float wb = w0[o * 258 + 129 + c]; a = wb; s = w0[o * 258 + c] + wb; }
  w2[i] = a; wsum[i] = s;
}

__global__ void k_zero(float* __restrict__ gsum, float* __restrict__ gsumsq) {
  int t = threadIdx.x;
  if (t < CC) { gsum[t] = 0.f; gsumsq[t] = 0.f; }
}

__global__ __launch_bounds__(128) void k_conv0(const float* __restrict__ featP,
                                               const int* __restrict__ idxb,
                                               const float* __restrict__ wsum,
                                               const float* __restrict__ w2,
                                               const float* __restrict__ bias,
                                               float* __restrict__ maxv, float* __restrict__ minv,
                                               float* __restrict__ gsum, float* __restrict__ gsumsq) {
  __shared__ float G[CP * KNB];    // gathered neighbor features [c][k]
  __shared__ float cent[CP];
  __shared__ float t0[CC];
  __shared__ float ob[CC * KNB];
  __shared__ int   nb[KNB];
  int n = blockIdx.x, b = blockIdx.y, tid = threadIdx.x;
  const float* fb = featP + (size_t)b * CP * NN;
  if (tid < KNB) nb[tid] = idxb[((size_t)b * NN + n) * KNB + tid];
  for (int c = tid; c < CP; c += 128) cent[c] = fb[(size_t)c * NN + n];
  __syncthreads();
  for (int t = tid; t < CP * KNB; t += 128) {
    int k = t / CP, c = t % CP;
    G[c * KNB + k] = fb[(size_t)c * NN + nb[k]];
  }
  __syncthreads();
  {
    float a = bias[tid];
    const float* wp = wsum + tid * CP;
    for (int c = 0; c < CP; ++c) a += wp[c] * cent[c];
    t0[tid] = a;
  }
  __syncthreads();
  int lane = tid & 31, wave = tid >> 5;
  int l15 = lane & 15, khalf = (lane >> 4) * 2, rofs = (lane < 16) ? 0 : 8;
  for (int j = 0; j < 4; ++j) {
    int job = wave * 4 + j, mt = job >> 1, nt = job & 1;
    int o = mt * 16 + l15, col = nt * 16 + l15;
    int cc = (col < KNB) ? col : 0;
    float msk = (col < KNB) ? 1.f : 0.f;
    v8f acc = {};
    for (int kc = 0; kc < 33; ++kc) {
      int kp = kc * 4 + khalf;
      v2f a, bm;
      a.x = w2[o * CP + kp];            a.y = w2[o * CP + kp + 1];
      bm.x = G[kp * KNB + cc] * msk;    bm.y = G[(kp + 1) * KNB + cc] * msk;
      acc = wmma4(a, bm, acc);
    }
    if (col < KNB) {
      for (int r = 0; r < 8; ++r) {
        int oo = mt * 16 + r + rofs;
        ob[oo * KNB + col] = t0[oo] - acc[r];
      }
    }
  }
  __syncthreads();
  {
    int o = tid;
    float mx = -3.4e38f, mn = 3.4e38f, s = 0.f, s2 = 0.f;
    for (int k = 0; k < KNB; ++k) {
      float v = ob[o * KNB + k];
      mx = fmaxf(mx, v); mn = fminf(mn, v); s += v; s2 += v * v;
    }
    size_t oi = (size_t)(b * CC + o) * NN + n;
    maxv[oi] = mx; minv[oi] = mn;
    atomicAdd(&gsum[o], s); atomicAdd(&gsumsq[o], s2);
  }
}

__global__ void k_bn0(const float* __restrict__ maxv, const float* __restrict__ minv,
                      const float* __restrict__ gsum, const float* __restrict__ gsumsq,
                      const float* __restrict__ g, const float* __restrict__ be,
                      float* __restrict__ X) {
  int blk = blockIdx.x, c = blk & 127;
  float mean = gsum[c] / CNT0;
  float var = gsumsq[c] / CNT0 - mean * mean;
  float a = g[c] * rsqrtf(var + 1e-5f);
  float cf = be[c] - a * mean;
  size_t base = (size_t)blk * NN;
  for (int n = threadIdx.x; n < NN; n += 256) {
    float v = (a >= 0.f) ? maxv[base + n] : minv[base + n];   // max(relu(a*x+cf)) over k
    X[base + n] = fmaxf(a * v + cf, 0.f);
  }
}

// ---------------- phase 4: res blocks (WMMA conv1x1 + fused IN/BN) ----------

__global__ __launch_bounds__(256) void k_resconv(const float* __restrict__ W,
                                                 const float* __restrict__ bias,
                                                 const float* __restrict__ IN,
                                                 float* __restrict__ OUT) {
  int b = blockIdx.y, nbase = blockIdx.x * 16;
  int tid = threadIdx.x, lane = tid & 31, wave = tid >> 5;
  int l15 = lane & 15, khalf = (lane >> 4) * 2, rofs = (lane < 16) ? 0 : 8;
  int o = wave * 16 + l15, n = nbase + l15;
  const float* inb = IN + (size_t)b * CC * NN;
  v8f acc = {};
  for (int kc = 0; kc < 32; ++kc) {
    int kp = kc * 4 + khalf;
    v2f a, bm;
    a.x = W[o * CC + kp];                 a.y = W[o * CC + kp + 1];
    bm.x = inb[(size_t)kp * NN + n];      bm.y = inb[(size_t)(kp + 1) * NN + n];
    acc = wmma4(a, bm, acc);
  }
  float* outb = OUT + (size_t)b * CC * NN;
  for (int r = 0; r < 8; ++r) {
    int oo = wave * 16 + r + rofs;
    outb[(size_t)oo * NN + n] = acc[r] + bias[oo];
  }
}

__global__ void k_stats(const float* __restrict__ T, float* __restrict__ meanA,
                        float* __restrict__ invA, float* __restrict__ ratioA) {
  int blk = blockIdx.x;                       // b*128 + c
  const float* p = T + (size_t)blk * NN;
  __shared__ float rs[256], rq[256];
  float s = 0.f, q2 = 0.f;
  for (int n = threadIdx.x; n < NN; n += 256) { float v = p[n]; s += v; q2 += v * v; }
  rs[threadIdx.x] = s; rq[threadIdx.x] = q2; __syncthreads();
  for (int st = 128; st > 0; st >>= 1) {
    if (threadIdx.x < st) { rs[threadIdx.x] += rs[threadIdx.x + st]; rq[threadIdx.x] += rq[threadIdx.x + st]; }
    __syncthreads();
  }
  if (threadIdx.x == 0) {
    float m = rs[0] / (float)NN;
    float var = rq[0] / (float)NN - m * m;
    meanA[blk] = m; invA[blk] = rsqrtf(var + 1e-5f); ratioA[blk] = var / (var + 1e-5f);
  }
}

__global__ void k_elem(const float* __restrict__ T, const float* __restrict__ meanA,
                       const float* __restrict__ invA, const float* __restrict__ ratioA,
                       const float* __restrict__ g, const float* __restrict__ be,
                       const float* __restrict__ skip, float* __restrict__ OUT, int mode) {
  int blk = blockIdx.x, c = blk & 127;
  float vb = 0.f;
  for (int bb = 0; bb < BB; ++bb) vb += ratioA[bb * CC + c];   // BN var = mean_b var/(var+eps)
  vb *= (1.f / (float)BB);
  float scale = g[c] * rsqrtf(vb + 1e-5f);
  float aa = invA[blk] * scale;
  float cf = be[c] - meanA[blk] * aa;
  size_t base = (size_t)blk * NN;
  for (int n = threadIdx.x; n < NN; n += 256) {
    float z = T[base + n] * aa + cf;
    if (mode) z += skip[base + n];
    OUT[base + n] = fmaxf(z, 0.f);
  }
}

__global__ void k_final(const float* __restrict__ X, const float* __restrict__ lw,
                        const float* __restrict__ lb, float* __restrict__ out) {
  int p = blockIdx.x * 256 + threadIdx.x;
  int b = p >> 11, n = p & 2047;
  const float* xp = X + (size_t)b * CC * NN + n;
  float s = lb[0];
  for (int c = 0; c < CC; ++c) s += lw[c] * xp[(size_t)c * NN];
  out[p] = s;
}

// ---------------- host orchestration ----------------------------------------

extern "C" void kernel_launch(void* const* d_in, const int* in_sizes, int n_in,
                              void* d_out, int out_size, void* d_ws, size_t ws_size,
                              hipStream_t stream) {
  const float* data   = (const float*)d_in[0];
  const float* c0w    = (const float*)d_in[2];
  const float* c0b    = (const float*)d_in[3];
  const float* bn0g   = (const float*)d_in[4];
  const float* bn0b   = (const float*)d_in[5];
  const float* linw   = (const float*)d_in[6];
  const float* linb   = (const float*)d_in[7];
  const float* rw1    = (const float*)d_in[8];
  const float* rb1    = (const float*)d_in[9];
  const float* rg1    = (const float*)d_in[10];
  const float* rbe1   = (const float*)d_in[11];
  const float* rw2    = (const float*)d_in[12];
  const float* rb2    = (const float*)d_in[13];
  const float* rg2    = (const float*)d_in[14];
  const float* rbe2   = (const float*)d_in[15];

  float* ws = (float*)d_ws;
  float* nrm    = ws;                    // 16384
  float* scol   = ws + 16384;            // 1024
  float* q      = ws + 17408;            // 16384
  float* gsum   = ws + 33792;            // 128
  float* gsumsq = ws + 33920;            // 128
  float* meanA  = ws + 34048;            // 1024
  float* invA   = ws + 35072;            // 1024
  float* ratioA = ws + 36096;            // 1024
  float* wsum   = ws + 37120;            // 16896
  float* w2     = ws + 54016;            // 16896
  int*   idxb   = (int*)(ws + 70912);    // 327680 ints
  float* featP  = ws + 398592;           // 2162688
  float* maxv   = ws + 2561280;          // 2097152
  float* minv   = ws + 4658432;          // 2097152
  float* X      = ws + 6755584;          // 2097152
  float* T      = ws + 8852736;          // 2097152
  float* H      = ws + 10949888;         // 2097152

  k_norms <<<64, 256, 0, stream>>>(data, nrm);
  k_colsum<<<1024, 256, 0, stream>>>(data, nrm, scol);
  k_feat  <<<64, 256, 0, stream>>>(data, scol, featP, q);
  k_knn   <<<dim3(128, 8), 128, 0, stream>>>(featP, q, idxb);
  k_prepw <<<66, 256, 0, stream>>>(c0w, wsum, w2);
  k_zero  <<<1, 256, 0, stream>>>(gsum, gsumsq);
  k_conv0 <<<dim3(2048, 8), 128, 0, stream>>>(featP, idxb, wsum, w2, c0b,
                                              maxv, minv, gsum, gsumsq);
  k_bn0   <<<1024, 256, 0, stream>>>(maxv, minv, gsum, gsumsq, bn0g, bn0b, X);

  for (int i = 0; i < 4; ++i) {
    k_resconv<<<dim3(128, 8), 256, 0, stream>>>(rw1 + (size_t)i * 16384, rb1 + i * 128, X, T);
    k_stats  <<<1024, 256, 0, stream>>>(T, meanA, invA, ratioA);
    k_elem   <<<1024, 256, 0, stream>>>(T, meanA, invA, ratioA, rg1 + i * 128, rbe1 + i * 128, X, H, 0);
    k_resconv<<<dim3(128, 8), 256, 0, stream>>>(rw2 + (size_t)i * 16384, rb2 + i * 128, H, T);
    k_stats  <<<1024, 256, 0, stream>>>(T, meanA, invA, ratioA);
    k_elem   <<<1024, 256, 0, stream>>>(T, meanA, invA, ratioA, rg2 + i * 128, rbe2 + i * 128, X, X, 1);
  }
  k_final<<<64, 256, 0, stream>>>(X, linw, linb, (float*)d_out);
}